// DMSA_31078383354473
// MI455X (gfx1250) — compile-verified
//
#include <hip/hip_runtime.h>
#include <math.h>

// ---------------- constants from the reference ----------------
#define B_      8
#define DIM_    384
#define H_      56
#define W_      56
#define HW_     (H_ * W_)          // 3136
#define DHEAD_  64
#define NHEAD_  (DIM_ / DHEAD_)    // 6
#define DG_     3                  // D_GROUPS
#define NGD_    (DIM_ / DG_)       // 128
#define HOFF_   28
#define WOFF_   28
#define L_      (HOFF_ * WOFF_)    // 784
#define SCALE_  0.125f             // 64^-0.5
#define BN_EPS_ 1e-06f

typedef __attribute__((ext_vector_type(2))) float v2f;
typedef __attribute__((ext_vector_type(8))) float v8f;

static __device__ __forceinline__ v8f wmma_f32(v2f a, v2f b, v8f c) {
    // V_WMMA_F32_16X16X4_F32 : D = A(16x4) * B(4x16) + C(16x16), fp32 throughout
    return __builtin_amdgcn_wmma_f32_16x16x4_f32(false, a, false, b, (short)0, c,
                                                 false, false);
}

static __device__ __forceinline__ float sigmoidf_(float x) {
    return 1.0f / (1.0f + __expf(-x));
}
static __device__ __forceinline__ float gelu_exact_(float x) {
    return 0.5f * x * (1.0f + erff(x * 0.70710678118654752440f));
}

// =====================================================================
// Kernel 1: q = einsum('bchw,oc->bohw', x, q_w)
// One wave -> one 16(O) x 64(pix) tile: A-fragment (weights) is reused by
// 4 WMMAs per k-step (2.5 loads/WMMA).  K=384 -> 96 k-steps, 384 WMMA/wave.
// q stored as q[((b*384+o)*3136)+pix]
// =====================================================================
__global__ __launch_bounds__(256) void k_qgemm(const float* __restrict__ x,
                                               const float* __restrict__ qw,
                                               float* __restrict__ q) {
    const int tid  = threadIdx.x;
    const int wave = (blockIdx.x * (blockDim.x >> 5)) + (tid >> 5);
    const int lane = tid & 31;
    const int lo   = lane & 15;
    const int hi   = lane >> 4;

    // tiles: 8 batches * 24 o-tiles * 49 pixel64-tiles
    const int PT = HW_ / 64;                 // 49
    const int b  = wave / (24 * PT);
    int rem      = wave - b * (24 * PT);
    const int ot = rem / PT;
    const int pt = rem - ot * PT;
    const int o0 = ot * 16;
    const int p0 = pt * 64;

    v8f acc[4] = {{}, {}, {}, {}};
    const float* xb = x + (size_t)b * DIM_ * HW_;
    #pragma unroll 4
    for (int kk = 0; kk < DIM_; kk += 4) {
        const int ka = kk + hi * 2;
        v2f a;
        a.x = qw[(o0 + lo) * DIM_ + ka];
        a.y = qw[(o0 + lo) * DIM_ + ka + 1];
        __builtin_prefetch(&xb[(size_t)(ka + 8) * HW_ + p0 + lo], 0, 1);
        #pragma unroll
        for (int t = 0; t < 4; ++t) {
            v2f bb;
            bb.x = xb[(size_t)(ka)     * HW_ + p0 + t * 16 + lo];
            bb.y = xb[(size_t)(ka + 1) * HW_ + p0 + t * 16 + lo];
            acc[t] = wmma_f32(a, bb, acc[t]);
        }
    }
    float* qb = q + (size_t)b * DIM_ * HW_;
    #pragma unroll
    for (int t = 0; t < 4; ++t)
        #pragma unroll
        for (int r = 0; r < 8; ++r) {
            const int o = o0 + r + 8 * hi;
            qb[(size_t)o * HW_ + p0 + t * 16 + lo] = acc[t][r];
        }
}

// =====================================================================
// Kernel 2: depthwise 5x5 stride-2 conv on q (grouped view) + bias + BN + GELU
// off_act[(n*128+c)*784 + pos],  n = b*3+g in [0,24)
// =====================================================================
__global__ __launch_bounds__(256) void k_dwconv(const float* __restrict__ q,
                                                const float* __restrict__ dww,
                                                const float* __restrict__ dwb,
                                                const float* __restrict__ bnw,
                                                const float* __restrict__ bnb,
                                                const float* __restrict__ bnm,
                                                const float* __restrict__ bnv,
                                                float* __restrict__ off_act) {
    const int idx = blockIdx.x * blockDim.x + threadIdx.x;   // 24*128*784
    const int pos = idx % L_;
    const int c   = (idx / L_) % NGD_;
    const int n   = idx / (L_ * NGD_);
    const int b   = n / DG_;
    const int g   = n - b * DG_;
    const int oy  = pos / WOFF_;
    const int ox  = pos - oy * WOFF_;

    const float* src = q + ((size_t)(b * DIM_ + g * NGD_ + c)) * HW_;
    const float* wv  = dww + c * 25;
    float acc = 0.0f;
    #pragma unroll
    for (int ky = 0; ky < 5; ++ky) {
        const int iy = oy * 2 - 2 + ky;
        if (iy < 0 || iy >= H_) continue;
        #pragma unroll
        for (int kx = 0; kx < 5; ++kx) {
            const int ix = ox * 2 - 2 + kx;
            if (ix < 0 || ix >= W_) continue;
            acc = fmaf(src[iy * W_ + ix], wv[ky * 5 + kx], acc);
        }
    }
    acc += dwb[c];
    const float sc = bnw[c] * __frsqrt_rn(bnv[c] + BN_EPS_);
    acc = (acc - bnm[c]) * sc + bnb[c];
    off_act[idx] = gelu_exact_(acc);
}

// =====================================================================
// Kernel 3: pointwise 128->3, tanh/sigmoid, grid build, bilinear sample,
// modulation.  One 128-thread block per (n, pos).
// x_s stored as x_s[((b*384+o)*784)+pos]
// =====================================================================
__global__ __launch_bounds__(128) void k_sample(const float* __restrict__ off_act,
                                                const float* __restrict__ pww,
                                                const float* __restrict__ x,
                                                float* __restrict__ xs) {
    __shared__ float red[3][128];
    __shared__ float sg[3];

    const int t   = threadIdx.x;          // channel within group
    const int blk = blockIdx.x;           // 24*784
    const int pos = blk % L_;
    const int n   = blk / L_;
    const int b   = n / DG_;
    const int g   = n - b * DG_;
    const int oy  = pos / WOFF_;
    const int ox  = pos - oy * WOFF_;

    const float ov = off_act[((size_t)(n * NGD_ + t)) * L_ + pos];
    red[0][t] = ov * pww[0 * NGD_ + t];
    red[1][t] = ov * pww[1 * NGD_ + t];
    red[2][t] = ov * pww[2 * NGD_ + t];
    __syncthreads();
    for (int s = 64; s > 0; s >>= 1) {
        if (t < s) {
            red[0][t] += red[0][t + s];
            red[1][t] += red[1][t + s];
            red[2][t] += red[2][t + s];
        }
        __syncthreads();
    }
    if (t == 0) {
        const float ry = (0.5f + (float)oy) / (float)HOFF_ * 2.0f - 1.0f;
        const float rx = (0.5f + (float)ox) / (float)WOFF_ * 2.0f - 1.0f;
        const float rngf = (1.0f / (float)HOFF_) * 2.0f;   // OFFSET_RANGE_FACTOR/28
        const float gy = tanhf(red[0][0]) * rngf + ry;     // pos[...,0] -> grid y
        const float gx = tanhf(red[1][0]) * rngf + rx;     // pos[...,1] -> grid x
        sg[0] = (gx + 1.0f) * 0.5f * (float)(W_ - 1);
        sg[1] = (gy + 1.0f) * 0.5f * (float)(H_ - 1);
        sg[2] = sigmoidf_(sigmoidf_(red[2][0]));  // reference applies sigmoid twice
    }
    __syncthreads();

    const float fx = sg[0], fy = sg[1], mod = sg[2];
    const float x0f = floorf(fx), y0f = floorf(fy);
    const float x1f = x0f + 1.0f, y1f = y0f + 1.0f;

    const float* img = x + ((size_t)(b * DIM_ + g * NGD_ + t)) * HW_;
    auto gather = [&](float yi, float xi) -> float {
        const bool valid = (yi >= 0.0f) && (yi <= (float)(H_ - 1)) &&
                           (xi >= 0.0f) && (xi <= (float)(W_ - 1));
        int yc = (int)fminf(fmaxf(yi, 0.0f), (float)(H_ - 1));
        int xc = (int)fminf(fmaxf(xi, 0.0f), (float)(W_ - 1));
        return valid ? img[yc * W_ + xc] : 0.0f;
    };
    float v = gather(y0f, x0f) * ((x1f - fx) * (y1f - fy))
            + gather(y1f, x0f) * ((x1f - fx) * (fy - y0f))
            + gather(y0f, x1f) * ((fx - x0f) * (y1f - fy))
            + gather(y1f, x1f) * ((fx - x0f) * (fy - y0f));
    xs[((size_t)(b * DIM_ + g * NGD_ + t)) * L_ + pos] = v * mod;
}

// =====================================================================
// Kernel 4: kv GEMM  (x_s -> K^T and V, attention-friendly layouts)
// One wave -> 64(O) x 16(pix): the B-fragment (activations) is reused by
// 4 WMMAs per k-step.
// kT[((b*6+h)*64 + d)*784 + l]   (d-major, l contiguous)
// v [((b*6+h)*784 + l)*64 + d]   (l-major)
// =====================================================================
__global__ __launch_bounds__(256) void k_kvgemm(const float* __restrict__ xs,
                                                const float* __restrict__ kvw,
                                                float* __restrict__ kT,
                                                float* __restrict__ v) {
    const int tid  = threadIdx.x;
    const int wave = (blockIdx.x * (blockDim.x >> 5)) + (tid >> 5);
    const int lane = tid & 31;
    const int lo   = lane & 15;
    const int hi   = lane >> 4;

    const int PT = L_ / 16;                       // 49
    const int b  = wave / (12 * PT);              // 12 o64-tiles
    int rem      = wave - b * (12 * PT);
    const int ot = rem / PT;
    const int pt = rem - ot * PT;
    const int o0 = ot * 64;
    const int p0 = pt * 16;

    v8f acc[4] = {{}, {}, {}, {}};
    const float* xb = xs + (size_t)b * DIM_ * L_;
    #pragma unroll 4
    for (int kk = 0; kk < DIM_; kk += 4) {
        const int ka = kk + hi * 2;
        v2f bb;
        bb.x = xb[(size_t)(ka)     * L_ + p0 + lo];
        bb.y = xb[(size_t)(ka + 1) * L_ + p0 + lo];
        __builtin_prefetch(&xb[(size_t)(ka + 8) * L_ + p0 + lo], 0, 1);
        #pragma unroll
        for (int t = 0; t < 4; ++t) {
            v2f a;
            a.x = kvw[(o0 + t * 16 + lo) * DIM_ + ka];
            a.y = kvw[(o0 + t * 16 + lo) * DIM_ + ka + 1];
            acc[t] = wmma_f32(a, bb, acc[t]);
        }
    }
    #pragma unroll
    for (int t = 0; t < 4; ++t)
        #pragma unroll
        for (int r = 0; r < 8; ++r) {
            int o = o0 + t * 16 + r + 8 * hi;
            const int pix = p0 + lo;
            if (o < DIM_) {                       // K half
                const int h = o / DHEAD_, d = o % DHEAD_;
                kT[((size_t)((b * NHEAD_ + h) * DHEAD_ + d)) * L_ + pix] = acc[t][r];
            } else {                              // V half
                o -= DIM_;
                const int h = o / DHEAD_, d = o % DHEAD_;
                v[(((size_t)(b * NHEAD_ + h) * L_) + pix) * DHEAD_ + d] = acc[t][r];
            }
        }
}

// =====================================================================
// Kernel 5: flash attention.  One wave = 16 query rows of one (b, head).
// S = Q K^T via WMMA (K-dim 64 -> 16 steps), online softmax,
// P tile through LDS (C-frag -> A-frag relayout), O += P V via WMMA.
// attn_out stored channel-major like q: ao[((b*384 + h*64 + d)*3136)+qpix]
// =====================================================================
__global__ __launch_bounds__(256) void k_attn(const float* __restrict__ q,
                                              const float* __restrict__ kT,
                                              const float* __restrict__ v,
                                              float* __restrict__ ao) {
    __shared__ float plds[8][16 * 16];
    const int tid  = threadIdx.x;
    const int wv   = tid >> 5;
    const int lane = tid & 31;
    const int lo   = lane & 15;
    const int hi   = lane >> 4;

    const int wave = blockIdx.x * 8 + wv;
    const int QT   = HW_ / 16;                   // 196
    const int b    = wave / (NHEAD_ * QT);
    int rem        = wave - b * (NHEAD_ * QT);
    const int h    = rem / QT;
    const int qt   = rem - h * QT;
    const int q0   = qt * 16;

    const float* qb  = q  + ((size_t)(b * DIM_ + h * DHEAD_)) * HW_;       // [d][pix]
    const float* ktb = kT + ((size_t)((b * NHEAD_ + h) * DHEAD_)) * L_;    // [d][l]
    const float* vb  = v  + ((size_t)(b * NHEAD_ + h) * L_) * DHEAD_;      // [l][d]

    // preload Q fragments: 16 k-steps covering d = 0..63
    v2f qf[16];
    #pragma unroll
    for (int s = 0; s < 16; ++s) {
        const int d = s * 4 + hi * 2;
        qf[s].x = qb[(size_t)d       * HW_ + q0 + lo];
        qf[s].y = qb[(size_t)(d + 1) * HW_ + q0 + lo];
    }

    v8f oacc[4] = {{}, {}, {}, {}};
    float mrow[8], lrow[8];
    #pragma unroll
    for (int r = 0; r < 8; ++r) { mrow[r] = -3.0e38f; lrow[r] = 0.0f; }

    for (int j = 0; j < L_ / 16; ++j) {          // 49 key tiles
        // ---- S = (Q K^T) * scale ----
        v8f s = {};
        #pragma unroll
        for (int ks = 0; ks < 16; ++ks) {
            v2f bb;
            const int d = ks * 4 + hi * 2;
            bb.x = ktb[(size_t)d       * L_ + j * 16 + lo];
            bb.y = ktb[(size_t)(d + 1) * L_ + j * 16 + lo];
            s = wmma_f32(qf[ks], bb, s);
        }
        // ---- online softmax + P tile to LDS ----
        #pragma unroll
        for (int r = 0; r < 8; ++r) {
            float cur = s[r] * SCALE_;
            float mx = cur;
            #pragma unroll
            for (int o = 1; o < 16; o <<= 1)
                mx = fmaxf(mx, __shfl_xor(mx, o, 16));
            const float nm   = fmaxf(mrow[r], mx);
            const float corr = __expf(mrow[r] - nm);
            const float p    = __expf(cur - nm);
            float ps = p;
            #pragma unroll
            for (int o = 1; o < 16; o <<= 1)
                ps += __shfl_xor(ps, o, 16);
            lrow[r] = lrow[r] * corr + ps;
            mrow[r] = nm;
            #pragma unroll
            for (int nt = 0; nt < 4; ++nt) oacc[nt][r] *= corr;
            plds[wv][(r + 8 * hi) * 16 + lo] = p;
        }
        asm volatile("s_wait_dscnt 0" ::: "memory");   // wave-local LDS RAW fence
        // ---- O += P V ----
        #pragma unroll
        for (int ks = 0; ks < 4; ++ks) {
            v2f a;
            const int kp = ks * 4 + hi * 2;
            a.x = plds[wv][lo * 16 + kp];
            a.y = plds[wv][lo * 16 + kp + 1];
            #pragma unroll
            for (int nt = 0; nt < 4; ++nt) {
                v2f bb;
                bb.x = vb[(size_t)(j * 16 + kp)     * DHEAD_ + nt * 16 + lo];
                bb.y = vb[(size_t)(j * 16 + kp + 1) * DHEAD_ + nt * 16 + lo];
                oacc[nt] = wmma_f32(a, bb, oacc[nt]);
            }
        }
        asm volatile("s_wait_dscnt 0" ::: "memory");   // drain before next overwrite
    }

    float* aob = ao + ((size_t)(b * DIM_ + h * DHEAD_)) * HW_;
    #pragma unroll
    for (int r = 0; r < 8; ++r) {
        const float inv = 1.0f / lrow[r];
        const int m = r + 8 * hi;
        #pragma unroll
        for (int nt = 0; nt < 4; ++nt)
            aob[(size_t)(nt * 16 + lo) * HW_ + q0 + m] = oacc[nt][r] * inv;
    }
}

// =====================================================================
// Kernel 6: proj GEMM + bias -> d_out  (layout (b, o, h, w))
// 16(O) x 64(pix) tiles like k_qgemm.
// =====================================================================
__global__ __launch_bounds__(256) void k_proj(const float* __restrict__ ao,
                                              const float* __restrict__ pw,
                                              const float* __restrict__ pb,
                                              float* __restrict__ out) {
    const int tid  = threadIdx.x;
    const int wave = (blockIdx.x * (blockDim.x >> 5)) + (tid >> 5);
    const int lane = tid & 31;
    const int lo   = lane & 15;
    const int hi   = lane >> 4;

    const int PT = HW_ / 64;                 // 49
    const int b  = wave / (24 * PT);
    int rem      = wave - b * (24 * PT);
    const int ot = rem / PT;
    const int pt = rem - ot * PT;
    const int o0 = ot * 16;
    const int p0 = pt * 64;

    v8f acc[4] = {{}, {}, {}, {}};
    const float* ab = ao + (size_t)b * DIM_ * HW_;
    #pragma unroll 4
    for (int kk = 0; kk < DIM_; kk += 4) {
        const int ka = kk + hi * 2;
        v2f a;
        a.x = pw[(o0 + lo) * DIM_ + ka];
        a.y = pw[(o0 + lo) * DIM_ + ka + 1];
        __builtin_prefetch(&ab[(size_t)(ka + 8) * HW_ + p0 + lo], 0, 1);
        #pragma unroll
        for (int t = 0; t < 4; ++t) {
            v2f bb;
            bb.x = ab[(size_t)(ka)     * HW_ + p0 + t * 16 + lo];
            bb.y = ab[(size_t)(ka + 1) * HW_ + p0 + t * 16 + lo];
            acc[t] = wmma_f32(a, bb, acc[t]);
        }
    }
    float* ob = out + (size_t)b * DIM_ * HW_;
    #pragma unroll
    for (int t = 0; t < 4; ++t)
        #pragma unroll
        for (int r = 0; r < 8; ++r) {
            const int o = o0 + r + 8 * hi;
            ob[(size_t)o * HW_ + p0 + t * 16 + lo] = acc[t][r] + pb[o];
        }
}

// =====================================================================
extern "C" void kernel_launch(void* const* d_in, const int* in_sizes, int n_in,
                              void* d_out, int out_size, void* d_ws, size_t ws_size,
                              hipStream_t stream) {
    const float* x    = (const float*)d_in[0];
    const float* q_w  = (const float*)d_in[1];
    const float* kv_w = (const float*)d_in[2];
    const float* pr_w = (const float*)d_in[3];
    const float* pr_b = (const float*)d_in[4];
    const float* dw_w = (const float*)d_in[5];
    const float* dw_b = (const float*)d_in[6];
    const float* bn_w = (const float*)d_in[7];
    const float* bn_b = (const float*)d_in[8];
    const float* bn_m = (const float*)d_in[9];
    const float* bn_v = (const float*)d_in[10];
    const float* pw_w = (const float*)d_in[11];
    float* out = (float*)d_out;

    float* ws = (float*)d_ws;
    const size_t SZ_Q   = (size_t)B_ * DIM_ * HW_;       // 9,633,792
    const size_t SZ_OFF = (size_t)B_ * DG_ * NGD_ * L_;  // 2,408,448
    const size_t SZ_XS  = (size_t)B_ * DIM_ * L_;        // 2,408,448
    float* qbuf = ws;                      ws += SZ_Q;
    float* offb = ws;                      ws += SZ_OFF;
    float* xsb  = ws;                      ws += SZ_XS;
    float* kTb  = ws;                      ws += SZ_XS;
    float* vbuf = ws;                      ws += SZ_XS;
    float* aob  = ws;                      ws += SZ_Q;

    // 1) q GEMM: 8 * 24 * 49 (16x64) tiles / 8 waves per block
    k_qgemm<<<(B_ * 24 * (HW_ / 64)) / 8, 256, 0, stream>>>(x, q_w, qbuf);
    // 2) depthwise + BN + GELU : 24*128*784 threads
    k_dwconv<<<(B_ * DG_ * NGD_ * L_) / 256, 256, 0, stream>>>(
        qbuf, dw_w, dw_b, bn_w, bn_b, bn_m, bn_v, offb);
    // 3) pointwise + grid + bilinear sample
    k_sample<<<B_ * DG_ * L_, 128, 0, stream>>>(offb, pw_w, x, xsb);
    // 4) kv GEMM: 8 * 12 * 49 (64x16) tiles / 8 waves per block
    k_kvgemm<<<(B_ * 12 * (L_ / 16)) / 8, 256, 0, stream>>>(xsb, kv_w, kTb, vbuf);
    // 5) attention: 8 * 6 * 196 waves / 8 per block
    k_attn<<<(B_ * NHEAD_ * (HW_ / 16)) / 8, 256, 0, stream>>>(qbuf, kTb, vbuf, aob);
    // 6) proj GEMM + bias
    k_proj<<<(B_ * 24 * (HW_ / 64)) / 8, 256, 0, stream>>>(aob, pr_w, pr_b, out);
}